// CPSFMemcellFusedReal_42872363548957
// MI455X (gfx1250) — compile-verified
//
#include <hip/hip_runtime.h>
#include <math.h>

// Problem constants (from the reference)
#define BB 128      // batch
#define NN 64       // latent dim
#define MM 16384    // memory slots
#define SS 64       // state dim
#define EPSV 1e-6f
#define MAXQ 25.0f
#define PI_F 3.14159265358979323846f
#define KS3 32      // split-K factor for long-K GEMMs

typedef __attribute__((ext_vector_type(2))) float v2f;
typedef __attribute__((ext_vector_type(8))) float v8f;

// fp32 WMMA: D(16x16,f32) = A(16x4,f32) * B(4x16,f32) + C
static __device__ __forceinline__ v8f wmma4(v2f a, v2f b, v8f c) {
  return __builtin_amdgcn_wmma_f32_16x16x4_f32(
      /*neg_a=*/false, a, /*neg_b=*/false, b,
      /*c_mod=*/(short)0, c, /*reuse_a=*/false, /*reuse_b=*/false);
}

// ---------------- workspace layout (floats) ----------------
constexpr size_t OFF_BHAT  = 0;                                   // M*N
constexpr size_t OFF_ZJN   = OFF_BHAT + (size_t)MM * NN;          // M
constexpr size_t OFF_ZJB   = OFF_ZJN + MM;                        // M
constexpr size_t OFF_WPERP = OFF_ZJB + MM;                        // M
constexpr size_t OFF_WDIFF = OFF_WPERP + MM;                      // M
constexpr size_t OFF_ZNORM = OFF_WDIFF + MM;                      // B
constexpr size_t OFF_GAIN  = OFF_ZNORM + BB;                      // B*M
constexpr size_t OFF_TB    = OFF_GAIN + (size_t)BB * MM;          // B*S
constexpr size_t OFF_E     = OFF_TB + (size_t)BB * SS;            // B*S
constexpr size_t OFF_GRAD  = OFF_E + (size_t)BB * SS;             // M*S
constexpr size_t OFF_PART  = OFF_GRAD + (size_t)MM * SS;          // KS3*B*S
constexpr size_t OFF_NSQ   = OFF_PART + (size_t)KS3 * BB * SS;    // M/4 = 4096 wave partials
constexpr size_t OFF_SCAL  = OFF_NSQ + 4096;                      // 1
// total ~4.6M floats (~18.4 MB) of d_ws

// ---------------- per-slot precompute: b_hat, ||z_j||^2, z_j.b, w_perp, w_diff
__global__ void k_pre_m(const float* __restrict__ zj, const float* __restrict__ vd,
                        const float* __restrict__ sp, const float* __restrict__ spp,
                        float* __restrict__ bhat, float* __restrict__ zjn,
                        float* __restrict__ zjb, float* __restrict__ wperp,
                        float* __restrict__ wdiff) {
  int wave = (int)((blockIdx.x * blockDim.x + threadIdx.x) >> 5);
  int lane = threadIdx.x & 31;
  if (wave >= MM) return;
  const float* vr = vd + (size_t)wave * NN;
  const float* zr = zj + (size_t)wave * NN;
  float v0 = vr[lane], v1 = vr[lane + 32];
  float z0 = zr[lane], z1 = zr[lane + 32];
  float dn2 = v0 * v0 + v1 * v1;
  float zn  = z0 * z0 + z1 * z1;
  float zb  = z0 * v0 + z1 * v1;
  #pragma unroll
  for (int off = 16; off; off >>= 1) {
    dn2 += __shfl_xor(dn2, off, 32);
    zn  += __shfl_xor(zn,  off, 32);
    zb  += __shfl_xor(zb,  off, 32);
  }
  float dn  = sqrtf(dn2);
  float inv = (dn > EPSV) ? (1.0f / dn) : 0.0f;
  bhat[(size_t)wave * NN + lane]      = v0 * inv;
  bhat[(size_t)wave * NN + 32 + lane] = v1 * inv;
  if (lane == 0) {
    zjn[wave] = zn;
    zjb[wave] = zb * inv;  // z_j . b_hat (0 if degenerate)
    const float tinyeps = 1.1920929e-07f;  // float32 eps
    float a = fmaxf(sp[wave], tinyeps), b = fmaxf(spp[wave], tinyeps);
    float wpar = 1.0f / (a * a), wpp = 1.0f / (b * b);
    wperp[wave] = wpp;
    wdiff[wave] = wpar - wpp;
  }
}

__global__ void k_znorm(const float* __restrict__ z, float* __restrict__ znorm) {
  int b = threadIdx.x;  // 128 threads
  float s = 0.f;
  for (int n = 0; n < NN; ++n) { float v = z[(size_t)b * NN + n]; s += v * v; }
  znorm[b] = s;
}

// ---------------- gain[B,M] via two WMMA-f32 GEMMs + elementwise exp
__global__ void k_gain(const float* __restrict__ z, const float* __restrict__ zj,
                       const float* __restrict__ bhat, const float* __restrict__ alphaj,
                       const float* __restrict__ zjn, const float* __restrict__ zjb,
                       const float* __restrict__ wperp, const float* __restrict__ wdiff,
                       const float* __restrict__ znorm, float* __restrict__ gain) {
  int gw   = (int)((blockIdx.x * blockDim.x + threadIdx.x) >> 5);
  int lane = threadIdx.x & 31;
  int half = lane >> 4, l = lane & 15;
  int mt = gw & ((MM / 16) - 1);  // 0..1023
  int bt = gw >> 10;              // 0..7
  const float* arow  = z    + (size_t)(bt * 16 + l) * NN + half * 2;
  const float* brow1 = zj   + (size_t)(mt * 16 + l) * NN + half * 2;
  const float* brow2 = bhat + (size_t)(mt * 16 + l) * NN + half * 2;
  v8f acc1 = {0.f,0.f,0.f,0.f,0.f,0.f,0.f,0.f};
  v8f acc2 = {0.f,0.f,0.f,0.f,0.f,0.f,0.f,0.f};
  #pragma unroll
  for (int kk = 0; kk < NN / 4; ++kk) {
    v2f a  = *(const v2f*)(arow  + kk * 4);
    v2f b1 = *(const v2f*)(brow1 + kk * 4);
    v2f b2 = *(const v2f*)(brow2 + kk * 4);
    acc1 = wmma4(a, b1, acc1);  // z @ z_j^T
    acc2 = wmma4(a, b2, acc2);  // z @ bhat^T
  }
  int m  = mt * 16 + l;
  float zjnm = zjn[m], zjbm = zjb[m], wpm = wperp[m], wdm = wdiff[m], alm = alphaj[m];
  #pragma unroll
  for (int r = 0; r < 8; ++r) {
    int brow = bt * 16 + r + half * 8;
    float dzn  = znorm[brow] - 2.0f * acc1[r] + zjnm;
    float proj = acc2[r] - zjbm;
    float q = fminf(wpm * dzn + wdm * proj * proj, MAXQ);
    gain[(size_t)brow * MM + m] = alm * expf(-PI_F * q);
  }
}

// ---------------- T_base partials: gain @ (T_hat + old_delta), split-K
__global__ void k_gemm3(const float* __restrict__ gain, const float* __restrict__ th,
                        const float* __restrict__ dl, float* __restrict__ part) {
  int gw   = (int)((blockIdx.x * blockDim.x + threadIdx.x) >> 5);  // 0..1023
  int lane = threadIdx.x & 31;
  int half = lane >> 4, l = lane & 15;
  int ks = gw & (KS3 - 1), tile = gw >> 5;
  int st = tile & 3, bt = tile >> 2;
  int kbase = ks * (MM / KS3);
  const float* arow = gain + (size_t)(bt * 16 + l) * MM + kbase + half * 2;
  int col = st * 16 + l;
  v8f acc = {0.f,0.f,0.f,0.f,0.f,0.f,0.f,0.f};
  for (int kk = 0; kk < (MM / KS3) / 4; ++kk) {
    int ka = kbase + kk * 4 + half * 2;
    v2f a = *(const v2f*)(arow + kk * 4);
    v2f b;
    b.x = th[(size_t)ka * SS + col]       + dl[(size_t)ka * SS + col];
    b.y = th[(size_t)(ka + 1) * SS + col] + dl[(size_t)(ka + 1) * SS + col];
    acc = wmma4(a, b, acc);
  }
  float* p = part + (size_t)ks * (BB * SS);
  #pragma unroll
  for (int r = 0; r < 8; ++r)
    p[(size_t)(bt * 16 + r + half * 8) * SS + col] = acc[r];
}

__global__ void k_reduce3(const float* __restrict__ part, const float* __restrict__ tstar,
                          float* __restrict__ tb, float* __restrict__ e) {
  int i = blockIdx.x * blockDim.x + threadIdx.x;  // 8192
  float s = 0.f;
  for (int ks = 0; ks < KS3; ++ks) s += part[(size_t)ks * (BB * SS) + i];
  tb[i] = s;
  e[i]  = s - tstar[i];
}

// ---------------- grad[M,S] = gain^T @ E  (+ per-wave ||.||^2 partial)
__global__ void k_gemm4(const float* __restrict__ gain, const float* __restrict__ e,
                        float* __restrict__ grad, float* __restrict__ nsqpart) {
  int gw   = (int)((blockIdx.x * blockDim.x + threadIdx.x) >> 5);  // 0..4095
  int lane = threadIdx.x & 31;
  int half = lane >> 4, l = lane & 15;
  int st = gw & 3, mt = gw >> 2;  // 0..1023
  int m   = mt * 16 + l;
  int col = st * 16 + l;
  v8f acc = {0.f,0.f,0.f,0.f,0.f,0.f,0.f,0.f};
  #pragma unroll 8
  for (int kk = 0; kk < BB / 4; ++kk) {
    int ka = kk * 4 + half * 2;
    v2f a, b;
    a.x = gain[(size_t)ka * MM + m];
    a.y = gain[(size_t)(ka + 1) * MM + m];
    b.x = e[(size_t)ka * SS + col];
    b.y = e[(size_t)(ka + 1) * SS + col];
    acc = wmma4(a, b, acc);
  }
  float nsq = 0.f;
  #pragma unroll
  for (int r = 0; r < 8; ++r) {
    float v = acc[r];
    grad[(size_t)(mt * 16 + r + half * 8) * SS + col] = v;
    nsq += v * v;
  }
  #pragma unroll
  for (int off = 16; off; off >>= 1) nsq += __shfl_xor(nsq, off, 32);
  if (lane == 0) nsqpart[gw] = nsq;
}

// ---------------- scalar: scale = alpha * min(DELTA_CAP/(alpha*||grad|| + tiny), 1)
__global__ void k_norm(const float* __restrict__ nsqpart, const float* __restrict__ alogit,
                       float* __restrict__ scal) {
  __shared__ float sm[256];
  float s = 0.f;
  for (int i = threadIdx.x; i < 4096; i += 256) s += nsqpart[i];
  sm[threadIdx.x] = s;
  __syncthreads();
  for (int off = 128; off; off >>= 1) {
    if ((int)threadIdx.x < off) sm[threadIdx.x] += sm[threadIdx.x + off];
    __syncthreads();
  }
  if (threadIdx.x == 0) {
    float alpha = 1.0f / (1.0f + expf(-alogit[0]));
    float n = alpha * sqrtf(sm[0]);
    float sc = fminf(1.0f / (n + 1.17549435e-38f), 1.0f);  // DELTA_CAP = 1
    scal[0] = alpha * sc;  // |delta| scale; sign applied in final kernel
  }
}

// ---------------- corr partials: gain @ grad, split-K (reuses part buffer)
__global__ void k_gemm5(const float* __restrict__ gain, const float* __restrict__ grad,
                        float* __restrict__ part) {
  int gw   = (int)((blockIdx.x * blockDim.x + threadIdx.x) >> 5);  // 0..1023
  int lane = threadIdx.x & 31;
  int half = lane >> 4, l = lane & 15;
  int ks = gw & (KS3 - 1), tile = gw >> 5;
  int st = tile & 3, bt = tile >> 2;
  int kbase = ks * (MM / KS3);
  const float* arow = gain + (size_t)(bt * 16 + l) * MM + kbase + half * 2;
  int col = st * 16 + l;
  v8f acc = {0.f,0.f,0.f,0.f,0.f,0.f,0.f,0.f};
  for (int kk = 0; kk < (MM / KS3) / 4; ++kk) {
    int ka = kbase + kk * 4 + half * 2;
    v2f a = *(const v2f*)(arow + kk * 4);
    v2f b;
    b.x = grad[(size_t)ka * SS + col];
    b.y = grad[(size_t)(ka + 1) * SS + col];
    acc = wmma4(a, b, acc);
  }
  float* p = part + (size_t)ks * (BB * SS);
  #pragma unroll
  for (int r = 0; r < 8; ++r)
    p[(size_t)(bt * 16 + r + half * 8) * SS + col] = acc[r];
}

// ---------------- out = T_base - scale * (gain @ grad)
__global__ void k_final(const float* __restrict__ part, const float* __restrict__ tb,
                        const float* __restrict__ scal, float* __restrict__ out) {
  int i = blockIdx.x * blockDim.x + threadIdx.x;  // 8192
  float s = 0.f;
  for (int ks = 0; ks < KS3; ++ks) s += part[(size_t)ks * (BB * SS) + i];
  out[i] = tb[i] - scal[0] * s;
}

extern "C" void kernel_launch(void* const* d_in, const int* in_sizes, int n_in,
                              void* d_out, int out_size, void* d_ws, size_t ws_size,
                              hipStream_t stream) {
  const float* z      = (const float*)d_in[0];  // [B,N]
  const float* tstar  = (const float*)d_in[1];  // [B,S]
  const float* zj     = (const float*)d_in[2];  // [M,N]
  const float* vd     = (const float*)d_in[3];  // [M,N]
  const float* th     = (const float*)d_in[4];  // [M,S]
  const float* dl     = (const float*)d_in[5];  // [M,S] old_delta
  const float* alphaj = (const float*)d_in[6];  // [M]
  const float* sp     = (const float*)d_in[7];  // [M]
  const float* spp    = (const float*)d_in[8];  // [M]
  const float* alogit = (const float*)d_in[9];  // scalar
  float* out = (float*)d_out;

  float* w = (float*)d_ws;
  float* bhat  = w + OFF_BHAT;
  float* zjn   = w + OFF_ZJN;
  float* zjb   = w + OFF_ZJB;
  float* wperp = w + OFF_WPERP;
  float* wdiff = w + OFF_WDIFF;
  float* znorm = w + OFF_ZNORM;
  float* gain  = w + OFF_GAIN;
  float* tb    = w + OFF_TB;
  float* e     = w + OFF_E;
  float* grad  = w + OFF_GRAD;
  float* part  = w + OFF_PART;
  float* nsqp  = w + OFF_NSQ;
  float* scal  = w + OFF_SCAL;

  // 1) per-slot constants + normalized directions (M waves)
  k_pre_m<<<MM / 8, 256, 0, stream>>>(zj, vd, sp, spp, bhat, zjn, zjb, wperp, wdiff);
  // 2) ||z_b||^2
  k_znorm<<<1, BB, 0, stream>>>(z, znorm);
  // 3) gain[B,M]: 8192 waves, WMMA f32
  k_gain<<<(BB / 16) * (MM / 16) / 8, 256, 0, stream>>>(z, zj, bhat, alphaj, zjn, zjb,
                                                        wperp, wdiff, znorm, gain);
  // 4) T_base split-K partials (32 tiles x KS3)
  k_gemm3<<<32 * KS3 / 8, 256, 0, stream>>>(gain, th, dl, part);
  k_reduce3<<<BB * SS / 256, 256, 0, stream>>>(part, tstar, tb, e);
  // 5) grad = gain^T @ E (4096 waves) + norm partials
  k_gemm4<<<(MM / 16) * (SS / 16) / 8, 256, 0, stream>>>(gain, e, grad, nsqp);
  k_norm<<<1, 256, 0, stream>>>(nsqp, alogit, scal);
  // 6) corr = gain @ grad partials, then out = T_base - scale*corr
  k_gemm5<<<32 * KS3 / 8, 256, 0, stream>>>(gain, grad, part);
  k_final<<<BB * SS / 256, 256, 0, stream>>>(part, tb, scal, out);
}